// TASTF_67078799228985
// MI455X (gfx1250) — compile-verified
//
#include <hip/hip_runtime.h>
#include <math.h>

// ---------------------------------------------------------------------------
// MI455X (gfx1250) implementation of the hetero-GNN + transformer forward.
// All large GEMMs run on V_WMMA_F32_16X16X4_F32 (fp32 WMMA, wave32 layouts per
// CDNA5 ISA 7.12.2).  Weights are staged in LDS (bank-conflict-free padded
// pitch), A-tiles are register-resident, B-fragments double-buffered so the
// WMMA pipe is not serialized behind memory latency.
// ---------------------------------------------------------------------------

typedef __attribute__((ext_vector_type(2))) float v2f;
typedef __attribute__((ext_vector_type(8))) float v8f;

#define B_    32
#define T_    24
#define N_    1000
#define NM_   200
#define NP_   400
#define NF_   400
#define D_    64
#define HID_  32
#define FFD_  128
#define NHEAD_ 4
#define HOR_  3
#define BN_   (B_ * N_)        // 32000
#define TOK_  (T_ * BN_)       // 768000

#define CDIV(a, b) (((a) + (b) - 1) / (b))

// ---------------------------------------------------------------------------
// WMMA GEMM:  Y[M, NT*16] = act( X[M,K] @ W[NT*16, K]^T + bias )
//  * one wave computes a 16 x (16*NT) strip; 8 waves / block
//  * W staged in LDS with pitch K+4 words: lane (lo,hi) reads word address
//    lo*(K+4) + 2*hi + k0  ->  bank 4*lo + 2*hi (mod 64): all 32 lanes hit
//    distinct bank pairs for the b64 read.
//  * A fragments (16x4 f32 layout: lane L<16 holds A[L][k0..k0+1],
//    lane L+16 holds A[L][k0+2..k0+3]) preloaded for the whole K.
//  * B fragments double-buffered across k-steps.
//  * C/D: VGPR r -> row r (lanes 0-15) / row r+8 (lanes 16-31), col = lane&15.
// ---------------------------------------------------------------------------
template <int NT, int K>
__global__ __launch_bounds__(256) void gemm_wmma_f32(
    const float* __restrict__ X, const float* __restrict__ W,
    const float* __restrict__ bias, float* __restrict__ Y,
    int M, int relu) {
  constexpr int N = NT * 16;
  constexpr int PITCH = K + 4;
  constexpr int KSTEPS = K / 4;
  __shared__ float wlds[N * PITCH];

  // cooperative staging of W (N x K, row-major, contiguous) into padded LDS
  for (int e = threadIdx.x * 4; e < N * K; e += 256 * 4) {
    const int row = e / K, col = e % K;  // col % 4 == 0 since K % 4 == 0
    const float4 w4 = *(const float4*)(W + e);
    *(float4*)(&wlds[row * PITCH + col]) = w4;
  }
  __syncthreads();

  const int lane = threadIdx.x & 31;
  const int wave = threadIdx.x >> 5;
  const int m0 = (blockIdx.x * 8 + wave) * 16;
  if (m0 >= M) return;  // after the only barrier; grids are exact anyway
  const int lo = lane & 15;
  const int hi = lane >> 4;  // 0 or 1

  // preload the whole 16 x K A tile for this wave (one clause of b64 loads)
  v2f a[KSTEPS];
  {
    const float* xrow = X + (size_t)(m0 + lo) * K + 2 * hi;
#pragma unroll
    for (int k = 0; k < KSTEPS; ++k) a[k] = *(const v2f*)(xrow + 4 * k);
  }

  v8f acc[NT];
#pragma unroll
  for (int n = 0; n < NT; ++n)
#pragma unroll
    for (int r = 0; r < 8; ++r) acc[n][r] = 0.0f;

  const float* wl = &wlds[lo * PITCH + 2 * hi];
  v2f bcur[NT], bnxt[NT];
#pragma unroll
  for (int n = 0; n < NT; ++n) bcur[n] = *(const v2f*)(wl + n * 16 * PITCH);

#pragma unroll
  for (int k = 0; k < KSTEPS; ++k) {
    if (k + 1 < KSTEPS) {
#pragma unroll
      for (int n = 0; n < NT; ++n)
        bnxt[n] = *(const v2f*)(wl + n * 16 * PITCH + 4 * (k + 1));
    }
#pragma unroll
    for (int n = 0; n < NT; ++n) {
      acc[n] = __builtin_amdgcn_wmma_f32_16x16x4_f32(
          /*neg_a=*/false, a[k], /*neg_b=*/false, bcur[n],
          /*c_mod=*/(short)0, acc[n], /*reuse_a=*/false, /*reuse_b=*/false);
    }
    if (k + 1 < KSTEPS) {
#pragma unroll
      for (int n = 0; n < NT; ++n) bcur[n] = bnxt[n];
    }
  }

#pragma unroll
  for (int n = 0; n < NT; ++n) {
    const int col = n * 16 + lo;
    const float bv = bias ? bias[col] : 0.0f;
#pragma unroll
    for (int r = 0; r < 8; ++r) {
      const int row = m0 + r + 8 * hi;
      float v = acc[n][r] + bv;
      if (relu) v = fmaxf(v, 0.0f);
      Y[(size_t)row * N + col] = v;
    }
  }
}

// ---------------------------------------------------------------------------
// Zero fill / small vector add
// ---------------------------------------------------------------------------
__global__ void fill_zero(float* __restrict__ p, size_t n) {
  size_t i = (size_t)blockIdx.x * blockDim.x + threadIdx.x;
  if (i < n) p[i] = 0.0f;
}

__global__ void vec_add2(const float* __restrict__ a, const float* __restrict__ b,
                         float* __restrict__ o, int n) {
  int i = blockIdx.x * blockDim.x + threadIdx.x;
  if (i < n) o[i] = a[i] + b[i];
}

// ---------------------------------------------------------------------------
// conv1 edge segment-sum over batch-0 sources (scalar feature) + counts
// ei: [2,E] int32 (row0 = src, row1 = dst)
// ---------------------------------------------------------------------------
__global__ void edge_sum1(const float* __restrict__ xseq, const int* __restrict__ ei,
                          int E, const int* __restrict__ src_gidx,
                          float* __restrict__ sums /*[T,ndst]*/,
                          float* __restrict__ cnt /*[ndst]*/, int ndst) {
  int id = blockIdx.x * blockDim.x + threadIdx.x;
  if (id >= E * T_) return;
  int e = id % E, t = id / E;
  int s = ei[e], d = ei[E + e];
  // x_src0[t, s] = x_seq[b=0, t, gidx[s]]
  float v = xseq[(size_t)t * N_ + src_gidx[s]];
  atomicAdd(&sums[(size_t)t * ndst + d], v);
  if (t == 0) atomicAdd(&cnt[d], 1.0f);
}

// ---------------------------------------------------------------------------
// conv1: out[t,b,i,o] = relu( x*WrA[o]+blA[o] (+x*WrB+blB)
//                            + (b==0)*(meanA*WlA[o] (+meanB*WlB[o])) )
// ---------------------------------------------------------------------------
__global__ void conv1_kernel(const float* __restrict__ xseq, const int* __restrict__ gidx,
                             int ndst,
                             const float* __restrict__ wrA, const float* __restrict__ blA,
                             const float* __restrict__ wlA, const float* __restrict__ sumA,
                             const float* __restrict__ cntA,
                             const float* __restrict__ wrB, const float* __restrict__ blB,
                             const float* __restrict__ wlB, const float* __restrict__ sumB,
                             const float* __restrict__ cntB,
                             float* __restrict__ out, int hasB) {
  int token = blockIdx.x * blockDim.x + threadIdx.x;
  int total = T_ * B_ * ndst;
  if (token >= total) return;
  int i = token % ndst;
  int tb = token / ndst;
  int b = tb % B_, t = tb / B_;
  float x = xseq[(size_t)b * T_ * N_ + (size_t)t * N_ + gidx[i]];
  float mA = 0.0f, mB = 0.0f;
  if (b == 0) {
    float cA = cntA[i];
    mA = (cA > 0.0f) ? sumA[(size_t)t * ndst + i] / cA : 0.0f;
    if (hasB) {
      float cB = cntB[i];
      mB = (cB > 0.0f) ? sumB[(size_t)t * ndst + i] / cB : 0.0f;
    }
  }
  float* op = out + (size_t)token * HID_;
#pragma unroll
  for (int o = 0; o < HID_; ++o) {
    float v = x * wrA[o] + blA[o];
    if (hasB) v += x * wrB[o] + blB[o];
    if (b == 0) {
      v += mA * wlA[o];
      if (hasB) v += mB * wlB[o];
    }
    op[o] = fmaxf(v, 0.0f);
  }
}

// ---------------------------------------------------------------------------
// conv2 edge segment-sum of batch-0 HID-dim features
// ---------------------------------------------------------------------------
__global__ void edge_sum2(const float* __restrict__ h1src /*[T,B,nsrc,HID]*/, int nsrc,
                          const int* __restrict__ ei, int E,
                          float* __restrict__ sums /*[T,ndst,HID]*/, int ndst) {
  int id = blockIdx.x * blockDim.x + threadIdx.x;
  if (id >= E * T_ * HID_) return;
  int c = id & (HID_ - 1);
  int et = id >> 5;
  int e = et % E, t = et / E;
  int s = ei[e], d = ei[E + e];
  float v = h1src[(((size_t)t * B_) * nsrc + s) * HID_ + c];  // batch 0
  atomicAdd(&sums[((size_t)t * ndst + d) * HID_ + c], v);
}

// ---------------------------------------------------------------------------
// conv2 batch-0 neighbor term: h2[t,0,i,o] += Wl @ mean (1 or 2 relations)
// ---------------------------------------------------------------------------
__global__ void conv2_b0add(float* __restrict__ h2 /*[T,B,ndst,D]*/, int ndst,
                            const float* __restrict__ wlA /*[D,HID]*/,
                            const float* __restrict__ sumA, const float* __restrict__ cntA,
                            const float* __restrict__ wlB,
                            const float* __restrict__ sumB, const float* __restrict__ cntB,
                            int hasB) {
  int id = blockIdx.x * blockDim.x + threadIdx.x;
  int total = T_ * ndst * D_;
  if (id >= total) return;
  int o = id & (D_ - 1);
  int ti = id >> 6;
  int i = ti % ndst, t = ti / ndst;
  float cA = cntA[i];
  float sA = (cA > 0.0f) ? 1.0f / cA : 0.0f;
  float accA = 0.0f;
  const float* sa = sumA + ((size_t)t * ndst + i) * HID_;
#pragma unroll
  for (int c = 0; c < HID_; ++c) accA += wlA[o * HID_ + c] * sa[c];
  float add = accA * sA;
  if (hasB) {
    float cB = cntB[i];
    float sB = (cB > 0.0f) ? 1.0f / cB : 0.0f;
    float accB = 0.0f;
    const float* sb = sumB + ((size_t)t * ndst + i) * HID_;
#pragma unroll
    for (int c = 0; c < HID_; ++c) accB += wlB[o * HID_ + c] * sb[c];
    add += accB * sB;
  }
  h2[(((size_t)t * B_) * ndst + i) * D_ + o] += add;  // batch 0 only
}

// ---------------------------------------------------------------------------
// token-parallel LayerNorm over D=64, optional residual add + optional relu
// out = [relu]( LN(a [+ add]) * g + b ); safe in-place (out == a)
// ---------------------------------------------------------------------------
__global__ void ln_kernel(const float* __restrict__ a, const float* __restrict__ addsrc,
                          const float* __restrict__ g, const float* __restrict__ bb,
                          float* __restrict__ out, int tokens, int has_add, int relu) {
  int t = blockIdx.x * blockDim.x + threadIdx.x;
  if (t >= tokens) return;
  const float* pa = a + (size_t)t * D_;
  const float* pd = addsrc ? addsrc + (size_t)t * D_ : nullptr;
  float buf[D_];
  float m = 0.0f;
#pragma unroll
  for (int i = 0; i < D_; ++i) {
    float v = pa[i];
    if (has_add) v += pd[i];
    buf[i] = v;
    m += v;
  }
  m *= (1.0f / D_);
  float var = 0.0f;
#pragma unroll
  for (int i = 0; i < D_; ++i) {
    float d = buf[i] - m;
    var += d * d;
  }
  var *= (1.0f / D_);
  float r = rsqrtf(var + 1e-5f);
  float* po = out + (size_t)t * D_;
#pragma unroll
  for (int i = 0; i < D_; ++i) {
    float v = (buf[i] - m) * r * g[i] + bb[i];
    if (relu) v = fmaxf(v, 0.0f);
    po[i] = v;
  }
}

// ---------------------------------------------------------------------------
// scatter projected node features into seq[(t, b*N+g), :] with +pos_enc(t, c)
// ---------------------------------------------------------------------------
__global__ void scatter_pos(const float* __restrict__ projout /*[T,B,num,D]*/,
                            const int* __restrict__ gidx, int num,
                            float* __restrict__ seq /*[T,B*N,D]*/) {
  int token = blockIdx.x * blockDim.x + threadIdx.x;
  int total = T_ * B_ * num;
  if (token >= total) return;
  int i = token % num;
  int tb = token / num;
  int b = tb % B_, t = tb / B_;
  int g = gidx[i];
  const float* sp = projout + (size_t)token * D_;
  float* dp = seq + (((size_t)t * B_ + b) * N_ + g) * D_;
  const float tf = (float)t;
#pragma unroll
  for (int c = 0; c < D_; ++c) {
    // div = exp(-(ln 1e4)/64 * (c & ~1)); even -> sin, odd -> cos
    float ang = tf * __expf((float)(c & ~1) * -0.14391157f);
    float pe = (c & 1) ? __cosf(ang) : __sinf(ang);
    dp[c] = sp[c] + pe;
  }
}

// ---------------------------------------------------------------------------
// Attention: one wave per (n, head).  K/V tiles staged in LDS.
// qkv layout: [(s*BN + n) * 192], q cols 0..63, k 64..127, v 128..191.
// attnO layout: [(s*BN + n) * 64 + h*16 + j]
// ---------------------------------------------------------------------------
__global__ __launch_bounds__(256) void attn_kernel(const float* __restrict__ qkv,
                                                   float* __restrict__ attnO) {
  __shared__ float kb[8][T_ * 16];
  __shared__ float vb[8][T_ * 16];
  const int lane = threadIdx.x & 31;
  const int wave = threadIdx.x >> 5;
  const int pair = blockIdx.x * 8 + wave;
  const bool active = pair < BN_ * NHEAD_;
  const int n = pair >> 2;
  const int h = pair & 3;

  if (active) {
    for (int i = lane; i < T_ * 16; i += 32) {
      int s = i >> 4, j = i & 15;
      size_t base = ((size_t)s * BN_ + n) * (3 * D_) + h * 16 + j;
      kb[wave][i] = qkv[base + D_];
      vb[wave][i] = qkv[base + 2 * D_];
    }
  }
  __syncthreads();

  if (!active || lane >= T_) return;
  const int s = lane;  // query row
  float q[16];
  {
    size_t base = ((size_t)s * BN_ + n) * (3 * D_) + h * 16;
#pragma unroll
    for (int j = 0; j < 16; ++j) q[j] = qkv[base + j];
  }
  float sc[T_];
  float mx = -3.4e38f;
#pragma unroll
  for (int t = 0; t < T_; ++t) {
    float d = 0.0f;
#pragma unroll
    for (int j = 0; j < 16; ++j) d += q[j] * kb[wave][t * 16 + j];
    d *= 0.25f;  // 1/sqrt(dh=16)
    sc[t] = d;
    mx = fmaxf(mx, d);
  }
  float sum = 0.0f;
#pragma unroll
  for (int t = 0; t < T_; ++t) {
    float e = __expf(sc[t] - mx);
    sc[t] = e;
    sum += e;
  }
  const float inv = 1.0f / sum;
  float o[16];
#pragma unroll
  for (int j = 0; j < 16; ++j) o[j] = 0.0f;
#pragma unroll
  for (int t = 0; t < T_; ++t) {
    float a = sc[t] * inv;
#pragma unroll
    for (int j = 0; j < 16; ++j) o[j] += a * vb[wave][t * 16 + j];
  }
  size_t base = ((size_t)s * BN_ + n) * D_ + h * 16;
#pragma unroll
  for (int j = 0; j < 16; ++j) attnO[base + j] = o[j];
}

// ---------------------------------------------------------------------------
// head: out[b, h, node] = seq[T-1, bn] . W[h] + b[h]
// ---------------------------------------------------------------------------
__global__ void head_kernel(const float* __restrict__ seq, const float* __restrict__ Wh,
                            const float* __restrict__ bh, float* __restrict__ out) {
  int id = blockIdx.x * blockDim.x + threadIdx.x;
  if (id >= BN_ * HOR_) return;
  int h = id % HOR_;
  int bn = id / HOR_;
  const float* xr = seq + (((size_t)(T_ - 1)) * BN_ + bn) * D_;
  float acc = bh[h];
#pragma unroll
  for (int c = 0; c < D_; ++c) acc += xr[c] * Wh[h * D_ + c];
  int b = bn / N_, node = bn % N_;
  out[((size_t)b * HOR_ + h) * N_ + node] = acc;
}

// ---------------------------------------------------------------------------
// Host side
// ---------------------------------------------------------------------------
// Input index map: assumes insertion-order flattening of setup_inputs():
//   0: x_seq
//   1..15 : conv1, rels {mm,pp,ff,mp,mf} x {Wl, bl, Wr}
//   16..30: conv2, same ordering
//   31: norm.g  32: norm.b
//   33..38: proj {macro, pico, femto} x {W, b}
//   39..62: tf layers 0..1 x {Wqkv,bqkv,Wo,bo,ln1g,ln1b,W1,b1,W2,b2,ln2g,ln2b}
//   63: head.W  64: head.b
//   65..69: e_mm, e_pp, e_ff, e_mp, e_mf     (int32 [2,E])
//   70..72: macro_idx, pico_idx, femto_idx   (int32)
enum { REL_MM = 0, REL_PP = 1, REL_FF = 2, REL_MP = 3, REL_MF = 4 };

extern "C" void kernel_launch(void* const* d_in, const int* in_sizes, int n_in,
                              void* d_out, int out_size, void* d_ws, size_t ws_size,
                              hipStream_t stream) {
  auto F = [&](int i) { return (const float*)d_in[i]; };
  auto I = [&](int i) { return (const int*)d_in[i]; };

  const float* x_seq = F(0);
  auto c1Wl = [&](int r) { return F(1 + r * 3 + 0); };
  auto c1bl = [&](int r) { return F(1 + r * 3 + 1); };
  auto c1Wr = [&](int r) { return F(1 + r * 3 + 2); };
  auto c2Wl = [&](int r) { return F(16 + r * 3 + 0); };
  auto c2bl = [&](int r) { return F(16 + r * 3 + 1); };
  auto c2Wr = [&](int r) { return F(16 + r * 3 + 2); };
  const float* norm_g = F(31);
  const float* norm_b = F(32);
  auto projW = [&](int ty) { return F(33 + ty * 2); };  // 0 macro, 1 pico, 2 femto
  auto projB = [&](int ty) { return F(34 + ty * 2); };
  auto tf = [&](int l, int k) { return F(39 + l * 12 + k); };
  const float* headW = F(63);
  const float* headB = F(64);
  const int* e_rel[5] = {I(65), I(66), I(67), I(68), I(69)};
  const int E_rel[5] = {in_sizes[65] / 2, in_sizes[66] / 2, in_sizes[67] / 2,
                        in_sizes[68] / 2, in_sizes[69] / 2};
  const int* macro_idx = I(70);
  const int* pico_idx = I(71);
  const int* femto_idx = I(72);

  float* ws = (float*)d_ws;
  // ---- workspace arenas (element offsets, 64-float aligned) ---------------
  auto rnd = [](size_t n) { return (n + 63) & ~(size_t)63; };
  size_t off = 0;
  const size_t SEQ = off;
  off += rnd((size_t)TOK_ * D_);
  const size_t ARENA = off;

  // graph-stage sub-arena (dead before transformer stage)
  size_t g = ARENA;
  auto ga = [&](size_t n) { size_t o = g; g += rnd(n); return o; };
  const size_t SUM1_BASE = g;
  size_t SUM1[5], CNT[5];
  const int NDST[5] = {NM_, NP_, NF_, NP_, NF_};
  for (int r = 0; r < 5; ++r) SUM1[r] = ga((size_t)T_ * NDST[r]);
  for (int r = 0; r < 5; ++r) CNT[r] = ga((size_t)NDST[r]);
  const size_t SUM1_LEN = g - SUM1_BASE;
  const size_t H1M = ga((size_t)T_ * B_ * NM_ * HID_);
  const size_t H1P = ga((size_t)T_ * B_ * NP_ * HID_);
  const size_t H1F = ga((size_t)T_ * B_ * NF_ * HID_);
  const size_t SUM2_BASE = g;
  size_t SUM2[5];
  for (int r = 0; r < 5; ++r) SUM2[r] = ga((size_t)T_ * NDST[r] * HID_);
  const size_t SUM2_LEN = g - SUM2_BASE;
  const size_t WRP = ga(D_ * HID_), BLP = ga(D_);
  const size_t WRF = ga(D_ * HID_), BLF = ga(D_);
  const size_t H2M = ga((size_t)T_ * B_ * NM_ * D_);
  const size_t H2P = ga((size_t)T_ * B_ * NP_ * D_);
  const size_t H2F = ga((size_t)T_ * B_ * NF_ * D_);
  const size_t PRM = ga((size_t)T_ * B_ * NM_ * D_);
  const size_t PRP = ga((size_t)T_ * B_ * NP_ * D_);
  const size_t PRF = ga((size_t)T_ * B_ * NF_ * D_);

  // transformer aliases onto the same arena
  const size_t QKV = ARENA;                       // TOK*192 (alias graph bufs)
  const size_t FFB = ARENA;                       // TOK*128 (alias qkv, qkv dead)
  size_t arena_end = ARENA + rnd((size_t)TOK_ * 3 * D_);
  if (g > arena_end) arena_end = g;
  const size_t ATT = arena_end;                   // TOK*64
  const size_t TMP = ATT + rnd((size_t)TOK_ * D_);

  const int BLK = 256;
  // ---- stage 1: conv1 segment sums + counts -------------------------------
  {
    size_t n = SUM1_LEN;
    fill_zero<<<CDIV(n, (size_t)BLK), BLK, 0, stream>>>(ws + SUM1_BASE, n);
  }
  const int* src_gidx[5] = {macro_idx, pico_idx, femto_idx, macro_idx, macro_idx};
  for (int r = 0; r < 5; ++r) {
    int nthr = E_rel[r] * T_;
    edge_sum1<<<CDIV(nthr, BLK), BLK, 0, stream>>>(x_seq, e_rel[r], E_rel[r], src_gidx[r],
                                                   ws + SUM1[r], ws + CNT[r], NDST[r]);
  }
  // ---- stage 2: conv1 + relu ----------------------------------------------
  {
    int nt = T_ * B_ * NM_;
    conv1_kernel<<<CDIV(nt, BLK), BLK, 0, stream>>>(
        x_seq, macro_idx, NM_, c1Wr(REL_MM), c1bl(REL_MM), c1Wl(REL_MM),
        ws + SUM1[REL_MM], ws + CNT[REL_MM], nullptr, nullptr, nullptr, nullptr, nullptr,
        ws + H1M, 0);
    nt = T_ * B_ * NP_;
    conv1_kernel<<<CDIV(nt, BLK), BLK, 0, stream>>>(
        x_seq, pico_idx, NP_, c1Wr(REL_PP), c1bl(REL_PP), c1Wl(REL_PP),
        ws + SUM1[REL_PP], ws + CNT[REL_PP], c1Wr(REL_MP), c1bl(REL_MP), c1Wl(REL_MP),
        ws + SUM1[REL_MP], ws + CNT[REL_MP], ws + H1P, 1);
    nt = T_ * B_ * NF_;
    conv1_kernel<<<CDIV(nt, BLK), BLK, 0, stream>>>(
        x_seq, femto_idx, NF_, c1Wr(REL_FF), c1bl(REL_FF), c1Wl(REL_FF),
        ws + SUM1[REL_FF], ws + CNT[REL_FF], c1Wr(REL_MF), c1bl(REL_MF), c1Wl(REL_MF),
        ws + SUM1[REL_MF], ws + CNT[REL_MF], ws + H1F, 1);
  }
  // ---- stage 3: conv2 segment sums ----------------------------------------
  {
    size_t n = SUM2_LEN;
    fill_zero<<<CDIV(n, (size_t)BLK), BLK, 0, stream>>>(ws + SUM2_BASE, n);
  }
  const size_t H1SRC[5] = {H1M, H1P, H1F, H1M, H1M};
  const int NSRC[5] = {NM_, NP_, NF_, NM_, NM_};
  for (int r = 0; r < 5; ++r) {
    int nthr = E_rel[r] * T_ * HID_;
    edge_sum2<<<CDIV(nthr, BLK), BLK, 0, stream>>>(ws + H1SRC[r], NSRC[r], e_rel[r],
                                                   E_rel[r], ws + SUM2[r], NDST[r]);
  }
  // ---- stage 4: conv2 (WMMA GEMM for Wr term; scalar add for batch-0 term)-
  vec_add2<<<CDIV(D_ * HID_, BLK), BLK, 0, stream>>>(c2Wr(REL_PP), c2Wr(REL_MP), ws + WRP, D_ * HID_);
  vec_add2<<<1, BLK, 0, stream>>>(c2bl(REL_PP), c2bl(REL_MP), ws + BLP, D_);
  vec_add2<<<CDIV(D_ * HID_, BLK), BLK, 0, stream>>>(c2Wr(REL_FF), c2Wr(REL_MF), ws + WRF, D_ * HID_);
  vec_add2<<<1, BLK, 0, stream>>>(c2bl(REL_FF), c2bl(REL_MF), ws + BLF, D_);
  {
    int Mm = T_ * B_ * NM_, Mp = T_ * B_ * NP_, Mf = T_ * B_ * NF_;
    gemm_wmma_f32<4, HID_><<<CDIV(Mm, 128), 256, 0, stream>>>(
        ws + H1M, c2Wr(REL_MM), c2bl(REL_MM), ws + H2M, Mm, 0);
    gemm_wmma_f32<4, HID_><<<CDIV(Mp, 128), 256, 0, stream>>>(
        ws + H1P, ws + WRP, ws + BLP, ws + H2P, Mp, 0);
    gemm_wmma_f32<4, HID_><<<CDIV(Mf, 128), 256, 0, stream>>>(
        ws + H1F, ws + WRF, ws + BLF, ws + H2F, Mf, 0);
  }
  {
    int nt = T_ * NM_ * D_;
    conv2_b0add<<<CDIV(nt, BLK), BLK, 0, stream>>>(ws + H2M, NM_, c2Wl(REL_MM),
        ws + SUM2[REL_MM], ws + CNT[REL_MM], nullptr, nullptr, nullptr, 0);
    nt = T_ * NP_ * D_;
    conv2_b0add<<<CDIV(nt, BLK), BLK, 0, stream>>>(ws + H2P, NP_, c2Wl(REL_PP),
        ws + SUM2[REL_PP], ws + CNT[REL_PP], c2Wl(REL_MP), ws + SUM2[REL_MP],
        ws + CNT[REL_MP], 1);
    nt = T_ * NF_ * D_;
    conv2_b0add<<<CDIV(nt, BLK), BLK, 0, stream>>>(ws + H2F, NF_, c2Wl(REL_FF),
        ws + SUM2[REL_FF], ws + CNT[REL_FF], c2Wl(REL_MF), ws + SUM2[REL_MF],
        ws + CNT[REL_MF], 1);
  }
  // ---- stage 5: norm + relu (in place), projection, scatter + pos enc -----
  {
    int tm = T_ * B_ * NM_, tp = T_ * B_ * NP_, tfk = T_ * B_ * NF_;
    ln_kernel<<<CDIV(tm, BLK), BLK, 0, stream>>>(ws + H2M, nullptr, norm_g, norm_b,
                                                 ws + H2M, tm, 0, 1);
    ln_kernel<<<CDIV(tp, BLK), BLK, 0, stream>>>(ws + H2P, nullptr, norm_g, norm_b,
                                                 ws + H2P, tp, 0, 1);
    ln_kernel<<<CDIV(tfk, BLK), BLK, 0, stream>>>(ws + H2F, nullptr, norm_g, norm_b,
                                                  ws + H2F, tfk, 0, 1);
    gemm_wmma_f32<4, D_><<<CDIV(tm, 128), 256, 0, stream>>>(
        ws + H2M, projW(0), projB(0), ws + PRM, tm, 0);
    gemm_wmma_f32<4, D_><<<CDIV(tp, 128), 256, 0, stream>>>(
        ws + H2P, projW(1), projB(1), ws + PRP, tp, 0);
    gemm_wmma_f32<4, D_><<<CDIV(tfk, 128), 256, 0, stream>>>(
        ws + H2F, projW(2), projB(2), ws + PRF, tfk, 0);
    scatter_pos<<<CDIV(tm, BLK), BLK, 0, stream>>>(ws + PRM, macro_idx, NM_, ws + SEQ);
    scatter_pos<<<CDIV(tp, BLK), BLK, 0, stream>>>(ws + PRP, pico_idx, NP_, ws + SEQ);
    scatter_pos<<<CDIV(tfk, BLK), BLK, 0, stream>>>(ws + PRF, femto_idx, NF_, ws + SEQ);
  }
  // ---- stage 6: transformer layers ----------------------------------------
  for (int l = 0; l < 2; ++l) {
    const float* Wqkv = tf(l, 0); const float* bqkv = tf(l, 1);
    const float* Wo = tf(l, 2);   const float* bo = tf(l, 3);
    const float* ln1g = tf(l, 4); const float* ln1b = tf(l, 5);
    const float* W1 = tf(l, 6);   const float* b1 = tf(l, 7);
    const float* W2 = tf(l, 8);   const float* b2 = tf(l, 9);
    const float* ln2g = tf(l, 10); const float* ln2b = tf(l, 11);

    gemm_wmma_f32<12, D_><<<CDIV(TOK_, 128), 256, 0, stream>>>(
        ws + SEQ, Wqkv, bqkv, ws + QKV, TOK_, 0);
    attn_kernel<<<CDIV(BN_ * NHEAD_, 8), 256, 0, stream>>>(ws + QKV, ws + ATT);
    gemm_wmma_f32<4, D_><<<CDIV(TOK_, 128), 256, 0, stream>>>(
        ws + ATT, Wo, bo, ws + TMP, TOK_, 0);
    ln_kernel<<<CDIV(TOK_, BLK), BLK, 0, stream>>>(ws + SEQ, ws + TMP, ln1g, ln1b,
                                                   ws + SEQ, TOK_, 1, 0);
    gemm_wmma_f32<8, D_><<<CDIV(TOK_, 128), 256, 0, stream>>>(
        ws + SEQ, W1, b1, ws + FFB, TOK_, 1);
    gemm_wmma_f32<4, FFD_><<<CDIV(TOK_, 128), 256, 0, stream>>>(
        ws + FFB, W2, b2, ws + TMP, TOK_, 0);
    ln_kernel<<<CDIV(TOK_, BLK), BLK, 0, stream>>>(ws + SEQ, ws + TMP, ln2g, ln2b,
                                                   ws + SEQ, TOK_, 1, 0);
  }
  // ---- stage 7: head ------------------------------------------------------
  head_kernel<<<CDIV(BN_ * HOR_, BLK), BLK, 0, stream>>>(ws + SEQ, headW, headB,
                                                         (float*)d_out);
  (void)n_in; (void)out_size; (void)ws_size;
}